// Model_28613072126388
// MI455X (gfx1250) — compile-verified
//
#include <hip/hip_runtime.h>

typedef float v2f __attribute__((ext_vector_type(2)));
typedef float v8f __attribute__((ext_vector_type(8)));

#define T_IN     80000
#define T_OUT    40000
#define NJ       (T_IN/2)  // float2 pairs per channel
#define MACRO    2048      // decimated outputs per macro-tile
#define NTHREADS 256       // 8 waves of 32
#define NWAVES   8
#define UELEN    2176      // >= MACRO + 65 phase samples
#define KSLICES  20        // K window of 80 per parity, 4 per WMMA

// ---- reference constants (fp32) ----
#define C1F        0.45244220f   // (2*SR - 2*pi*f)/(2*SR + 2*pi*f)
#define C2F        0.27377890f   // (1 - C1)/2
#define IHC_KF     1225.0f
#define ATAN_BETA  -0.78539816f  // atan(-1)
#define INV_NORM   0.42441318f   // 1/(3*pi/4)
#define P1F        34.657359f    // ln(2^50 - 1)
#define PPI_SCALE  0.017312341f  // P3/P1 = 0.012/ln2
#define CI0        4166.6667f    // SPONT/PIREST
#define CL0        5000.0f       // CI0*(PIREST+PL)/PL
// step (matches reference update order: cl uses the NEW ci):
//   a   = 0.994 - 0.1*ppi
//   ci' = a*ci + 0.006*cl
//   cl' = 0.0006*ci' + 0.9991*cl + 2.0
#define KA_D   0.994f
#define KA_P   0.1f
#define K_CICL 0.006f
#define K_CLCI 0.0006f
#define K_CLCL 0.9991f
#define K_CLB  2.0f

__device__ __forceinline__ float ihc_nl(float x) {
    return (atanf(fmaf(IHC_KF, x, -1.0f)) - ATAN_BETA) * INV_NORM;
}

// affine map s -> M*s + v over (ci, cl)
struct Aff { float m00, m01, m10, m11, v0, v1; };

__device__ __forceinline__ Aff aff_compose(const Aff& f, const Aff& g) {
    // apply g first, then f
    Aff r;
    r.m00 = f.m00 * g.m00 + f.m01 * g.m10;
    r.m01 = f.m00 * g.m01 + f.m01 * g.m11;
    r.m10 = f.m10 * g.m00 + f.m11 * g.m10;
    r.m11 = f.m10 * g.m01 + f.m11 * g.m11;
    r.v0  = f.m00 * g.v0  + f.m01 * g.v1 + f.v0;
    r.v1  = f.m10 * g.v0  + f.m11 * g.v1 + f.v1;
    return r;
}

__device__ __forceinline__ Aff aff_shfl_up(const Aff& a, int d) {
    Aff r;
    r.m00 = __shfl_up(a.m00, d); r.m01 = __shfl_up(a.m01, d);
    r.m10 = __shfl_up(a.m10, d); r.m11 = __shfl_up(a.m11, d);
    r.v0  = __shfl_up(a.v0,  d); r.v1  = __shfl_up(a.v1,  d);
    return r;
}

__global__ __launch_bounds__(NTHREADS) void cochlea_fused_kernel(
        const float* __restrict__ x, float* __restrict__ out, int nchan)
{
    __shared__ float her[96];            // even taps, reversed: her[79-t] = h[2t]
    __shared__ float hor[96];            // odd taps,  reversed: hor[79-t] = h[2t+1]
    __shared__ float ue[UELEN];          // even-phase nonlinearity
    __shared__ float uo[UELEN];          // odd-phase nonlinearity
    __shared__ float ppi[MACRO];         // rate function per macro-tile
    __shared__ float wT[NWAVES][6];      // per-wave scan totals
    __shared__ float wS[NWAVES][6];      // inclusive wave prefixes
    __shared__ float carry[2];           // (ci, cl) across macro-tiles

    const int ch = blockIdx.x;
    if (ch >= nchan) return;
    const float2* __restrict__ x2 = (const float2*)(x + (long long)ch * T_IN);
    float* __restrict__ oc = out + (long long)ch * T_OUT;

    const int tid  = threadIdx.x;
    const int lane = tid & 31;
    const int wv   = tid >> 5;
    const int hi   = lane >> 4;
    const int lo   = lane & 15;

    // ---- FIR taps of the 7-pole cascade (exact IIR impulse, 128 taps), reversed ----
    for (int i = tid; i < 96; i += NTHREADS) { her[i] = 0.0f; hor[i] = 0.0f; }
    __syncthreads();
    if (tid == 0) {
        float xp[7], yp[7];
        #pragma unroll
        for (int i = 0; i < 7; ++i) { xp[i] = 0.0f; yp[i] = 0.0f; }
        for (int n = 0; n < 128; ++n) {
            float cur = (n == 0) ? 1.0f : 0.0f;
            #pragma unroll
            for (int i = 0; i < 7; ++i) {
                float y = C2F * (cur + xp[i]) + C1F * yp[i];
                xp[i] = cur; yp[i] = y; cur = y;
            }
            int t = n >> 1;
            if (n & 1) hor[79 - t] = cur; else her[79 - t] = cur;
        }
        carry[0] = CI0; carry[1] = CL0;
    }
    __syncthreads();

    // ---- hoist all A-matrix WMMA slices into registers (constant for whole kernel) ----
    // A[m, k_abs] = h_tap[m + 63 - k_abs]; lanes 0-15 hold K={0,1}, lanes 16-31 K={2,3}
    v2f aE[KSLICES], aO[KSLICES];
    #pragma unroll
    for (int s = 0; s < KSLICES; ++s) {
        int ja = 16 - lo + 4 * s + 2 * hi;
        aE[s][0] = her[ja]; aE[s][1] = her[ja + 1];
        aO[s][0] = hor[ja]; aO[s][1] = hor[ja + 1];
    }

    for (int base = 0; base < T_OUT; base += MACRO) {
        // ---- phase-split nonlinearity: ue[i]=u[2*(base-64+i)], uo[i]=u[2*(base-64+i)+1] ----
        for (int idx = tid; idx < MACRO + 65; idx += NTHREADS) {
            int j = base - 64 + idx;
            float ve = 0.0f, vo = 0.0f;
            if (j >= 0 && j < NJ) {
                float2 w = x2[j];
                ve = ihc_nl(w.x);
                vo = ihc_nl(w.y);
            }
            ue[idx] = ve; uo[idx] = vo;
        }
        // prefetch next macro-tile's input while we compute this one
        {
            int pj = base + MACRO - 64 + tid * 8;
            if (pj >= 0 && pj < NJ) __builtin_prefetch(&x2[pj], 0, 0);
        }
        __syncthreads();

        // ---- FIR-decimate via WMMA: wave wv computes D[m,n] = d[base + 256*wv + 16n + m] ----
        {
            const int bw = 256 * wv;
            v8f acc = {0.0f,0.0f,0.0f,0.0f,0.0f,0.0f,0.0f,0.0f};
            #pragma unroll
            for (int s = 0; s < KSLICES; ++s) {
                const int ks = 4 * s;
                v2f b;
                const int ib = bw + 1 + 16 * lo + ks + 2 * hi;   // even phase (+1 shift)
                b[0] = ue[ib]; b[1] = ue[ib + 1];
                acc = __builtin_amdgcn_wmma_f32_16x16x4_f32(false, aE[s], false, b,
                                                            (short)0, acc, false, false);
                const int ibo = bw + 16 * lo + ks + 2 * hi;      // odd phase
                b[0] = uo[ibo]; b[1] = uo[ibo + 1];
                acc = __builtin_amdgcn_wmma_f32_16x16x4_f32(false, aO[s], false, b,
                                                            (short)0, acc, false, false);
            }
            #pragma unroll
            for (int j = 0; j < 8; ++j) {
                int m = j + 8 * hi;
                float d = acc[j];
                ppi[bw + 16 * lo + m] = PPI_SCALE * log1pf(expf(P1F * d));
            }
        }
        __syncthreads();

        // ---- adaptation: 8 samples/thread, shuffle-based affine scan ----
        Aff cur; cur.m00 = 1; cur.m01 = 0; cur.m10 = 0; cur.m11 = 1; cur.v0 = 0; cur.v1 = 0;
        const int g0 = base + 8 * tid;
        float pv[8];
        #pragma unroll
        for (int i = 0; i < 8; ++i) {
            int g = g0 + i;
            float p = ppi[8 * tid + i];
            pv[i] = p;
            if (g >= 1 && g < T_OUT) {
                float a = KA_D - KA_P * p;
                float n00 = a * cur.m00 + K_CICL * cur.m10;
                float n01 = a * cur.m01 + K_CICL * cur.m11;
                float n10 = K_CLCI * n00 + K_CLCL * cur.m10;
                float n11 = K_CLCI * n01 + K_CLCL * cur.m11;
                float w0  = a * cur.v0 + K_CICL * cur.v1;
                float w1  = K_CLCI * w0 + K_CLCL * cur.v1 + K_CLB;
                cur.m00 = n00; cur.m01 = n01; cur.m10 = n10; cur.m11 = n11;
                cur.v0 = w0; cur.v1 = w1;
            }
        }
        // intra-wave inclusive scan (no barriers)
        Aff inc = cur;
        #pragma unroll
        for (int d = 1; d < 32; d <<= 1) {
            Aff o = aff_shfl_up(inc, d);
            if (lane >= d) inc = aff_compose(inc, o);
        }
        if (lane == 31) {
            wT[wv][0] = inc.m00; wT[wv][1] = inc.m01; wT[wv][2] = inc.m10;
            wT[wv][3] = inc.m11; wT[wv][4] = inc.v0;  wT[wv][5] = inc.v1;
        }
        __syncthreads();
        if (wv == 0 && lane < NWAVES) {
            Aff t;
            t.m00 = wT[lane][0]; t.m01 = wT[lane][1]; t.m10 = wT[lane][2];
            t.m11 = wT[lane][3]; t.v0 = wT[lane][4];  t.v1 = wT[lane][5];
            #pragma unroll
            for (int d = 1; d < NWAVES; d <<= 1) {
                Aff o = aff_shfl_up(t, d);
                if (lane >= d) t = aff_compose(t, o);
            }
            wS[lane][0] = t.m00; wS[lane][1] = t.m01; wS[lane][2] = t.m10;
            wS[lane][3] = t.m11; wS[lane][4] = t.v0;  wS[lane][5] = t.v1;
        }
        __syncthreads();

        // exclusive prefix = (intra-wave exclusive) ∘ (previous-wave inclusive)
        float c0 = carry[0], c1 = carry[1];
        Aff we; we.m00 = 1; we.m01 = 0; we.m10 = 0; we.m11 = 1; we.v0 = 0; we.v1 = 0;
        if (wv > 0) {
            we.m00 = wS[wv-1][0]; we.m01 = wS[wv-1][1]; we.m10 = wS[wv-1][2];
            we.m11 = wS[wv-1][3]; we.v0 = wS[wv-1][4];  we.v1 = wS[wv-1][5];
        }
        Aff ie = aff_shfl_up(inc, 1);
        if (lane == 0) { ie.m00 = 1; ie.m01 = 0; ie.m10 = 0; ie.m11 = 1; ie.v0 = 0; ie.v1 = 0; }
        Aff E = aff_compose(ie, we);
        float ci = E.m00 * c0 + E.m01 * c1 + E.v0;
        float cl = E.m10 * c0 + E.m11 * c1 + E.v1;

        // replay & store (threads own 8 consecutive outputs)
        float r[8];
        #pragma unroll
        for (int i = 0; i < 8; ++i) {
            float p = pv[i];
            float a   = KA_D - KA_P * p;
            float nci = a * ci + K_CICL * cl;
            float ncl = K_CLCI * nci + K_CLCL * cl + K_CLB;
            int g = g0 + i;
            if (g >= 1 && g < T_OUT) { ci = nci; cl = ncl; }
            r[i] = nci * p;
        }
        if (g0 == 0) r[0] = 50.0f;  // SPONT at t=0
        if (g0 + 8 <= T_OUT) {
            float4 o0 = make_float4(r[0], r[1], r[2], r[3]);
            float4 o1 = make_float4(r[4], r[5], r[6], r[7]);
            *(float4*)(oc + g0)     = o0;
            *(float4*)(oc + g0 + 4) = o1;
        } else {
            for (int i = 0; i < 8; ++i)
                if (g0 + i < T_OUT) oc[g0 + i] = r[i];
        }
        __syncthreads();
        if (tid == 0) {
            carry[0] = wS[NWAVES-1][0] * c0 + wS[NWAVES-1][1] * c1 + wS[NWAVES-1][4];
            carry[1] = wS[NWAVES-1][2] * c0 + wS[NWAVES-1][3] * c1 + wS[NWAVES-1][5];
        }
        __syncthreads();
    }
}

extern "C" void kernel_launch(void* const* d_in, const int* in_sizes, int n_in,
                              void* d_out, int out_size, void* d_ws, size_t ws_size,
                              hipStream_t stream) {
    const float* x = (const float*)d_in[0];
    float* out = (float*)d_out;
    const int nchan = in_sizes[0] / T_IN;   // 4*60 = 240
    cochlea_fused_kernel<<<nchan, NTHREADS, 0, stream>>>(x, out, nchan);
}